// FrameMaskedContrastLoss2_75694503625206
// MI455X (gfx1250) — compile-verified
//
#include <hip/hip_runtime.h>
#include <math.h>

// Problem constants (match the reference)
#define MARGIN_ 0.99f
#define B_ 32
#define D_ 128
#define T_ 4096

typedef __attribute__((ext_vector_type(2))) float v2f;
typedef __attribute__((ext_vector_type(8))) float v8f;

// One wave (32 lanes) handles one tile of 16 consecutive t-values for one b.
// The D=128 reduction is done with 32 chained V_WMMA_F32_16X16X4_F32 ops:
//   A[m,k] = diff^2[d0+k, t0+m],  B = ones  =>  D[m,n] = sum_k diff^2[k, t0+m]
// (every column of D identical, so only the A and C/D register layouts matter).
__global__ __launch_bounds__(256) void frame_loss_partial_kernel(
    const float* __restrict__ pred1,
    const float* __restrict__ pred2,
    const int*   __restrict__ labels,
    const int*   __restrict__ n_frames,
    float*       __restrict__ partial)
{
    __shared__ float wsum[8];

    const int wave = threadIdx.x >> 5;
    const int lane = threadIdx.x & 31;
    const int tile = blockIdx.x * 8 + wave;      // 0 .. 8191
    const int b    = tile >> 8;                  // T_/16 = 256 tiles per batch row
    const int t0   = (tile & 255) << 4;          // first of 16 t-values
    const int m    = lane & 15;                  // A-matrix row index held by this lane
    const int half = lane >> 4;                  // 0: K={0,1}, 1: K={2,3}

    const long long base = (long long)b * (D_ * T_) + (t0 + m);

    v8f c = {};                                  // fp32 accumulator, 16x16
    const v2f bones = {1.0f, 1.0f};              // B = ones (layout-invariant)

    #pragma unroll 4
    for (int k0 = 0; k0 < D_; k0 += 4) {
        const int d0 = k0 + 2 * half;
        const float* p1 = pred1 + base + (long long)d0 * T_;
        const float* p2 = pred2 + base + (long long)d0 * T_;

        const float x0 = p1[0]  - p2[0];         // d = d0   (lanes<16: k0, lanes>=16: k0+2)
        const float x1 = p1[T_] - p2[T_];        // d = d0+1

        v2f a;
        a.x = x0 * x0;
        a.y = x1 * x1;

        if (k0 + 8 < D_) {                       // gfx1250 global_prefetch_b8, 8 d-rows ahead
            __builtin_prefetch(p1 + 8ll * T_, 0, 1);
            __builtin_prefetch(p2 + 8ll * T_, 0, 1);
        }

        // D = A x B + C  (v_wmma_f32_16x16x4_f32)
        c = __builtin_amdgcn_wmma_f32_16x16x4_f32(
                /*neg_a=*/false, a, /*neg_b=*/false, bones,
                /*c_mod=*/(short)0, c, /*reuse_a=*/false, /*reuse_b=*/false);
    }

    // Lane holds full sum-over-D for t = t0 + v + 8*half in c[v] (16-way replicated).
    const int   n  = n_frames[b];
    const float nf = (float)n;
    float acc = 0.0f;
    #pragma unroll
    for (int v = 0; v < 8; ++v) {
        const int   t  = t0 + v + 8 * half;
        const float s  = c[v];
        const float dv = sqrtf(s) * nf;                 // torch.dist(...) * frame
        const float mg = fmaxf(MARGIN_ - dv, 0.0f);
        const float per = (labels[b * T_ + t] != 0) ? (dv * dv) : (mg * mg);
        acc += (t < n) ? per : 0.0f;
    }
    acc *= (1.0f / 16.0f);                              // undo 16-lane replication

    // Deterministic wave32 tree reduction
    #pragma unroll
    for (int off = 16; off > 0; off >>= 1)
        acc += __shfl_xor(acc, off, 32);

    if (lane == 0) wsum[wave] = acc;
    __syncthreads();
    if (threadIdx.x == 0) {
        float s = 0.0f;
        #pragma unroll
        for (int w = 0; w < 8; ++w) s += wsum[w];
        partial[blockIdx.x] = s;                        // fixed-order, deterministic
    }
}

// Tiny deterministic finalize: sum 1024 partials in fixed order, divide by mask sum.
__global__ void frame_loss_finalize_kernel(const float* __restrict__ partial,
                                           int nparts,
                                           const int* __restrict__ n_frames,
                                           float* __restrict__ out)
{
    if (threadIdx.x == 0 && blockIdx.x == 0) {
        float s = 0.0f;
        for (int i = 0; i < nparts; ++i) s += partial[i];
        float cnt = 0.0f;
        for (int b = 0; b < B_; ++b) cnt += (float)n_frames[b];
        out[0] = s / cnt;
    }
}

extern "C" void kernel_launch(void* const* d_in, const int* in_sizes, int n_in,
                              void* d_out, int out_size, void* d_ws, size_t ws_size,
                              hipStream_t stream) {
    const float* pred1    = (const float*)d_in[0];
    const float* pred2    = (const float*)d_in[1];
    const int*   labels   = (const int*)d_in[2];
    const int*   n_frames = (const int*)d_in[3];
    float*       out      = (float*)d_out;
    float*       partial  = (float*)d_ws;               // 1024 floats = 4 KB scratch

    const int ntiles  = (B_ * T_) / 16;                 // 8192 wave-tiles
    const int nblocks = ntiles / 8;                     // 1024 blocks, 8 waves each

    frame_loss_partial_kernel<<<nblocks, 256, 0, stream>>>(
        pred1, pred2, labels, n_frames, partial);
    frame_loss_finalize_kernel<<<1, 32, 0, stream>>>(
        partial, nblocks, n_frames, out);
}